// Model_11751030522070
// MI455X (gfx1250) — compile-verified
//
#include <hip/hip_runtime.h>
#include <hip/hip_bf16.h>
#include <math.h>

#define LATDIM     64
#define NREL       32
#define RES_LAMBDA 0.5f
#define LEAKY      0.2f
#define EPSF       1e-12f

typedef __attribute__((ext_vector_type(2))) float v2f;
typedef __attribute__((ext_vector_type(8))) float v8f;

// ---- order-preserving float <-> uint mapping (for atomicMax on floats) ----
__device__ __forceinline__ unsigned f2ord(float f) {
    unsigned u = __float_as_uint(f);
    return (u & 0x80000000u) ? ~u : (u | 0x80000000u);
}
__device__ __forceinline__ float ord2f(unsigned u) {
    u = (u & 0x80000000u) ? (u & 0x7FFFFFFFu) : ~u;
    return __uint_as_float(u);
}

// ---------------------------------------------------------------------------
// ent = E0 ; res = E0
__global__ void copy_init_kernel(const float* __restrict__ E0,
                                 float* __restrict__ ent,
                                 float* __restrict__ res, int n) {
    int i = blockIdx.x * blockDim.x + threadIdx.x;
    if (i >= n) return;
    float v = E0[i];
    ent[i] = v;
    res[i] = v;
}

// ---------------------------------------------------------------------------
// Vcat[k][j] (64x64): j<32 -> W[k,:]·rel[j]  (head half of v_j)
//                     j>=32 -> W[64+k,:]·rel[j-32]  (tail half)
__global__ void vcat_kernel(const float* __restrict__ W,
                            const float* __restrict__ rel,
                            float* __restrict__ vcat) {
    int idx = blockIdx.x * blockDim.x + threadIdx.x; // 0..4095
    if (idx >= 64 * 64) return;
    int k = idx & 63;
    int j = idx >> 6;
    int r    = (j < NREL) ? j : (j - NREL);
    int wrow = (j < NREL) ? k : (64 + k);
    const float* wr = W + wrow * LATDIM;
    const float* rr = rel + r * LATDIM;
    float acc = 0.0f;
#pragma unroll
    for (int c = 0; c < LATDIM; ++c) acc = fmaf(wr[c], rr[c], acc);
    vcat[k * 64 + j] = acc;
}

// ---------------------------------------------------------------------------
// S[n][j] = sum_k ent[n][k] * Vcat[k][j]   via V_WMMA_F32_16X16X4_F32
// Block = 256 threads = 8 waves; each wave owns a 16-row strip, all 4 N-tiles.
__global__ __launch_bounds__(256)
void score_gemm_wmma(const float* __restrict__ ent,
                     const float* __restrict__ vcat,
                     float* __restrict__ S, int nStrips) {
    __shared__ float sB[64 * 64];        // Vcat, 16 KB
    __shared__ float sA[8][16 * 64];     // one 16x64 strip per wave, 32 KB

    const int tid  = threadIdx.x;
    const int wave = tid >> 5;
    const int lane = tid & 31;

    // Stage Vcat cooperatively (coalesced).
#pragma unroll
    for (int i = 0; i < 16; ++i) sB[tid + 256 * i] = vcat[tid + 256 * i];

    // Stage this wave's A strip (16 rows x 64 cols = 256 float4).
    const int strip   = blockIdx.x * 8 + wave;
    const bool valid  = strip < nStrips;
    const int rowBase = strip * 16;
    {
        const float4  z    = {0.f, 0.f, 0.f, 0.f};
        const float4* src  = (const float4*)(ent + (size_t)rowBase * 64);
        float4*       dstA = (float4*)(sA[wave]);
#pragma unroll
        for (int i = 0; i < 8; ++i) {
            int o = lane + 32 * i;
            dstA[o] = valid ? src[o] : z;
        }
    }
    __syncthreads();
    if (!valid) return;

    const int m  = lane & 15;         // A row in tile / output col in tile
    const int kb = (lane >> 4) * 2;   // which K-pair this half-wave holds

#pragma unroll
    for (int tn = 0; tn < 4; ++tn) {
        const int tileN = tn * 16;
        v8f c = {};
#pragma unroll
        for (int ks = 0; ks < 16; ++ks) {
            const int k0 = ks * 4 + kb;
            v2f a, b;
            a.x = sA[wave][m * 64 + k0];
            a.y = sA[wave][m * 64 + k0 + 1];
            b.x = sB[(k0)     * 64 + tileN + m];
            b.y = sB[(k0 + 1) * 64 + tileN + m];
            // 8 args: (neg_a, A, neg_b, B, c_mod, C, reuse_a, reuse_b)
            c = __builtin_amdgcn_wmma_f32_16x16x4_f32(
                    false, a, false, b, (short)0, c, false, false);
        }
        // C layout: VGPR i holds M=i (lanes 0-15) and M=8+i (lanes 16-31)
        const int col = tileN + m;
#pragma unroll
        for (int i = 0; i < 8; ++i) {
            const int row = rowBase + ((lane >> 4) * 8) + i;
            S[(size_t)row * 64 + col] = c[i];
        }
    }
}

// ---------------------------------------------------------------------------
__global__ void zero_kernel(unsigned* __restrict__ mbits,
                            float* __restrict__ denom,
                            float* __restrict__ agg,
                            int nNodes, int aggElems) {
    int i = blockIdx.x * blockDim.x + threadIdx.x;
    if (i < aggElems) agg[i] = 0.0f;
    if (i < nNodes) { mbits[i] = 0u; denom[i] = 0.0f; }
}

// ---------------------------------------------------------------------------
// per-edge logit + leaky relu + segment max (as ordered-uint atomicMax)
__global__ void logit_kernel(const float* __restrict__ S,
                             const int* __restrict__ ei,
                             const int* __restrict__ et,
                             float* __restrict__ elog,
                             unsigned* __restrict__ mbits, int nEdges) {
    int e = blockIdx.x * blockDim.x + threadIdx.x;
    if (e >= nEdges) return;
    int h = ei[e];
    int t = ei[nEdges + e];
    int r = et[e];
    float x = S[(size_t)h * 64 + r] + S[(size_t)t * 64 + NREL + r];
    x = (x < 0.0f) ? LEAKY * x : x;
    elog[e] = x;
    atomicMax(&mbits[h], f2ord(x));
}

// ---------------------------------------------------------------------------
// ex = exp(e - m[head]); denom[head] += ex
__global__ void exp_kernel(const float* __restrict__ elog,
                           const int* __restrict__ ei,
                           const unsigned* __restrict__ mbits,
                           float* __restrict__ ex,
                           float* __restrict__ denom, int nEdges) {
    int e = blockIdx.x * blockDim.x + threadIdx.x;
    if (e >= nEdges) return;
    int h = ei[e];
    float v = __expf(elog[e] - ord2f(mbits[h]));
    ex[e] = v;
    atomicAdd(&denom[h], v);
}

// ---------------------------------------------------------------------------
// agg[h] += (ex/denom[h]) * ent[t]   (64 lanes per edge)
__global__ void agg_kernel(const float* __restrict__ ent,
                           const int* __restrict__ ei,
                           const float* __restrict__ ex,
                           const float* __restrict__ denom,
                           float* __restrict__ agg, int nEdges) {
    long long gid = (long long)blockIdx.x * blockDim.x + threadIdx.x;
    int e = (int)(gid >> 6);
    int d = (int)(gid & 63);
    if (e >= nEdges) return;
    int h = ei[e];
    int t = ei[nEdges + e];
    float alpha = ex[e] / denom[h];
    atomicAdd(&agg[(size_t)h * 64 + d], alpha * ent[(size_t)t * 64 + d]);
}

// ---------------------------------------------------------------------------
// ent = normalize(agg + ent); res = 0.5*res + ent   (one wave32 per row)
__global__ __launch_bounds__(256)
void finalize_kernel(float* __restrict__ ent,
                     const float* __restrict__ agg,
                     float* __restrict__ res, int nNodes) {
    int wave = threadIdx.x >> 5;
    int lane = threadIdx.x & 31;
    int row  = blockIdx.x * 8 + wave;
    if (row >= nNodes) return;
    size_t base = (size_t)row * 64;
    float v0 = agg[base + lane]      + ent[base + lane];
    float v1 = agg[base + lane + 32] + ent[base + lane + 32];
    float ss = v0 * v0 + v1 * v1;
#pragma unroll
    for (int off = 16; off > 0; off >>= 1) ss += __shfl_xor(ss, off, 32);
    float inv = 1.0f / fmaxf(sqrtf(ss), EPSF);
    float n0 = v0 * inv, n1 = v1 * inv;
    ent[base + lane]      = n0;
    ent[base + lane + 32] = n1;
    res[base + lane]      = RES_LAMBDA * res[base + lane]      + n0;
    res[base + lane + 32] = RES_LAMBDA * res[base + lane + 32] + n1;
}

// ---------------------------------------------------------------------------
extern "C" void kernel_launch(void* const* d_in, const int* in_sizes, int n_in,
                              void* d_out, int out_size, void* d_ws, size_t ws_size,
                              hipStream_t stream) {
    const float* E0  = (const float*)d_in[0];   // entity_emb  [N,64]
    const float* R   = (const float*)d_in[1];   // relation_emb[32,64]
    const float* W   = (const float*)d_in[2];   // W [128,64]
    const int*   ei  = (const int*)d_in[3];     // edge_index [2,E]
    const int*   et  = (const int*)d_in[4];     // edge_type  [E]

    const int nNodes = in_sizes[0] / LATDIM;    // 200000
    const int nEdges = in_sizes[4];             // 1200000
    float* res = (float*)d_out;

    // carve workspace (256B aligned)
    char* p = (char*)d_ws;
    auto carve = [&](size_t bytes) -> void* {
        void* q = (void*)p;
        p += (bytes + 255) & ~(size_t)255;
        return q;
    };
    float*    ent   = (float*)   carve((size_t)nNodes * 64 * sizeof(float));
    float*    S     = (float*)   carve((size_t)nNodes * 64 * sizeof(float));
    float*    agg   = (float*)   carve((size_t)nNodes * 64 * sizeof(float));
    unsigned* mbits = (unsigned*)carve((size_t)nNodes * sizeof(unsigned));
    float*    denom = (float*)   carve((size_t)nNodes * sizeof(float));
    float*    elog  = (float*)   carve((size_t)nEdges * sizeof(float));
    float*    exv   = (float*)   carve((size_t)nEdges * sizeof(float));
    float*    vcat  = (float*)   carve((size_t)64 * 64 * sizeof(float));

    const int totE    = nNodes * 64;
    const int nStrips = nNodes / 16;            // 200000 is an exact multiple of 16

    copy_init_kernel<<<(totE + 255) / 256, 256, 0, stream>>>(E0, ent, res, totE);

    for (int hop = 0; hop < 2; ++hop) {
        vcat_kernel<<<16, 256, 0, stream>>>(W, R, vcat);
        score_gemm_wmma<<<(nStrips + 7) / 8, 256, 0, stream>>>(ent, vcat, S, nStrips);
        zero_kernel<<<(totE + 255) / 256, 256, 0, stream>>>(mbits, denom, agg, nNodes, totE);
        logit_kernel<<<(nEdges + 255) / 256, 256, 0, stream>>>(S, ei, et, elog, mbits, nEdges);
        exp_kernel<<<(nEdges + 255) / 256, 256, 0, stream>>>(elog, ei, mbits, exv, denom, nEdges);
        long long aggThreads = (long long)nEdges * 64;
        agg_kernel<<<(unsigned)((aggThreads + 255) / 256), 256, 0, stream>>>(ent, ei, exv, denom, agg, nEdges);
        finalize_kernel<<<(nNodes + 7) / 8, 256, 0, stream>>>(ent, agg, res, nNodes);
    }
}